// dCov2_30855045054965
// MI455X (gfx1250) — compile-verified
//
#include <hip/hip_runtime.h>
#include <math.h>

typedef float v2f __attribute__((ext_vector_type(2)));
typedef float v8f __attribute__((ext_vector_type(8)));

#define N_ROWS 8192
#define DX 512
#define DY 256

// ---------------------------------------------------------------------------
// Workspace layout (floats):
//   [0..1]  : double Sab accumulator
//   [2..3]  : pad
//   [4          .. 4+8192)   : Rx  (row sums of Dx)
//   [4+8192    .. 4+16384)   : Ry  (row sums of Dy)
//   [4+16384   .. 4+24576)   : sqx (||x_i||^2)
//   [4+24576   .. 4+32768)   : sqy (||y_i||^2)
// ---------------------------------------------------------------------------

__global__ void dcov_zero(float* ws) {
    int i = blockIdx.x * 256 + threadIdx.x;
    if (i < 4 + 2 * N_ROWS) ws[i] = 0.0f;   // Sab(+pad) + Rx + Ry
}

// One wave per row: squared L2 norm.
__global__ void row_sqnorm(const float* __restrict__ src, float* __restrict__ dst, int D) {
    const int lane = threadIdx.x & 31;
    const int wid  = threadIdx.x >> 5;
    const int row  = blockIdx.x * 8 + wid;
    const float* p = src + (size_t)row * D;
    float acc = 0.0f;
    for (int c = lane; c < D; c += 32) {
        float v = p[c];
        acc = fmaf(v, v, acc);
    }
    acc += __shfl_xor(acc, 1);
    acc += __shfl_xor(acc, 2);
    acc += __shfl_xor(acc, 4);
    acc += __shfl_xor(acc, 8);
    acc += __shfl_xor(acc, 16);
    if (lane == 0) dst[row] = acc;
}

// Main fused kernel: each wave owns a 32x32 super-tile (2x2 blocking of
// 16x16 WMMA tiles) of BOTH Gram matrices; epilogue converts to distances,
// accumulates Sab, Rx, Ry.
__global__ void __launch_bounds__(256)
dcov_tile(const float* __restrict__ x, const float* __restrict__ y,
          const float* __restrict__ sqx, const float* __restrict__ sqy,
          float* __restrict__ Rx, float* __restrict__ Ry,
          double* __restrict__ sab) {
    const int lane = threadIdx.x & 31;
    const int wid  = threadIdx.x >> 5;
    const int half = lane >> 4;          // 0: lanes 0-15, 1: lanes 16-31
    const int l16  = lane & 15;
    const int I0 = blockIdx.y * 32;
    const int J0 = (blockIdx.x * 8 + wid) * 32;
    const int kb = half * 2;             // per-lane k base within a 4-wide K slab

    // ---- Gram tiles for x (K = 512) ----
    v8f gx[2][2] = {};
    {
        const float* a0 = x + (size_t)(I0 + l16) * DX + kb;
        const float* a1 = a0 + (size_t)16 * DX;
        const float* b0 = x + (size_t)(J0 + l16) * DX + kb;
        const float* b1 = b0 + (size_t)16 * DX;
        for (int k = 0; k < DX; k += 4) {
            v2f A0 = *(const v2f*)(a0 + k);
            v2f A1 = *(const v2f*)(a1 + k);
            v2f B0 = *(const v2f*)(b0 + k);
            v2f B1 = *(const v2f*)(b1 + k);
            gx[0][0] = __builtin_amdgcn_wmma_f32_16x16x4_f32(false, A0, false, B0, (short)0, gx[0][0], false, false);
            gx[0][1] = __builtin_amdgcn_wmma_f32_16x16x4_f32(false, A0, false, B1, (short)0, gx[0][1], false, false);
            gx[1][0] = __builtin_amdgcn_wmma_f32_16x16x4_f32(false, A1, false, B0, (short)0, gx[1][0], false, false);
            gx[1][1] = __builtin_amdgcn_wmma_f32_16x16x4_f32(false, A1, false, B1, (short)0, gx[1][1], false, false);
        }
    }

    // ---- Gram tiles for y (K = 256) ----
    v8f gy[2][2] = {};
    {
        const float* a0 = y + (size_t)(I0 + l16) * DY + kb;
        const float* a1 = a0 + (size_t)16 * DY;
        const float* b0 = y + (size_t)(J0 + l16) * DY + kb;
        const float* b1 = b0 + (size_t)16 * DY;
        for (int k = 0; k < DY; k += 4) {
            v2f A0 = *(const v2f*)(a0 + k);
            v2f A1 = *(const v2f*)(a1 + k);
            v2f B0 = *(const v2f*)(b0 + k);
            v2f B1 = *(const v2f*)(b1 + k);
            gy[0][0] = __builtin_amdgcn_wmma_f32_16x16x4_f32(false, A0, false, B0, (short)0, gy[0][0], false, false);
            gy[0][1] = __builtin_amdgcn_wmma_f32_16x16x4_f32(false, A0, false, B1, (short)0, gy[0][1], false, false);
            gy[1][0] = __builtin_amdgcn_wmma_f32_16x16x4_f32(false, A1, false, B0, (short)0, gy[1][0], false, false);
            gy[1][1] = __builtin_amdgcn_wmma_f32_16x16x4_f32(false, A1, false, B1, (short)0, gy[1][1], false, false);
        }
    }

    // ---- Epilogue: Gram -> distances, reductions ----
    // C layout (ISA 7.12.2): reg r, lane L -> row m = r + 8*(L>>4), col n = L&15.
    float pr = 0.0f;
#pragma unroll
    for (int ti = 0; ti < 2; ++ti) {
        const int Ib = I0 + 16 * ti;
#pragma unroll
        for (int tj = 0; tj < 2; ++tj) {
            const int Jb = J0 + 16 * tj;
            const float sqxj = sqx[Jb + l16];
            const float sqyj = sqy[Jb + l16];
#pragma unroll
            for (int r = 0; r < 8; ++r) {
                const int i = Ib + r + 8 * half;
                float d2x = sqx[i] + sqxj - 2.0f * gx[ti][tj][r];
                float d2y = sqy[i] + sqyj - 2.0f * gy[ti][tj][r];
                float dx = (d2x > 0.0f) ? sqrtf(d2x) : 0.0f;
                float dy = (d2y > 0.0f) ? sqrtf(d2y) : 0.0f;
                pr = fmaf(dx, dy, pr);
                // row-sum over the 16 columns (xor masks 1..8 stay within
                // each 16-lane half)
                float sx = dx, sy = dy;
                sx += __shfl_xor(sx, 1); sx += __shfl_xor(sx, 2);
                sx += __shfl_xor(sx, 4); sx += __shfl_xor(sx, 8);
                sy += __shfl_xor(sy, 1); sy += __shfl_xor(sy, 2);
                sy += __shfl_xor(sy, 4); sy += __shfl_xor(sy, 8);
                if (l16 == r) {          // lanes r and 16+r own rows Ib+r / Ib+r+8
                    atomicAdd(&Rx[i], sx);
                    atomicAdd(&Ry[i], sy);
                }
            }
        }
    }

    // Block-level reduction of the elementwise-product sum, one f64 atomic/block.
    pr += __shfl_xor(pr, 1);  pr += __shfl_xor(pr, 2);
    pr += __shfl_xor(pr, 4);  pr += __shfl_xor(pr, 8);
    pr += __shfl_xor(pr, 16);
    __shared__ float red[8];
    if (lane == 0) red[wid] = pr;
    __syncthreads();
    if (threadIdx.x == 0) {
        float t = 0.0f;
        for (int w = 0; w < 8; ++w) t += red[w];
        atomicAdd(sab, (double)t);
    }
}

// Final combine: out = Sab/N^2 - 2*sum(Rx*Ry)/N^3 + sum(Rx)*sum(Ry)/N^4
__global__ void dcov_final(const float* __restrict__ Rx, const float* __restrict__ Ry,
                           const double* __restrict__ sab, float* __restrict__ out) {
    const int tid = threadIdx.x;  // 256 threads
    double srr = 0.0, sx = 0.0, sy = 0.0;
    for (int i = tid; i < N_ROWS; i += 256) {
        double a = Rx[i], b = Ry[i];
        srr += a * b;
        sx += a;
        sy += b;
    }
    __shared__ double s0[256], s1[256], s2[256];
    s0[tid] = srr; s1[tid] = sx; s2[tid] = sy;
    __syncthreads();
    for (int off = 128; off > 0; off >>= 1) {
        if (tid < off) {
            s0[tid] += s0[tid + off];
            s1[tid] += s1[tid + off];
            s2[tid] += s2[tid + off];
        }
        __syncthreads();
    }
    if (tid == 0) {
        const double N = (double)N_ROWS;
        double r = sab[0] / (N * N)
                 - 2.0 * s0[0] / (N * N * N)
                 + s1[0] * s2[0] / (N * N * N * N);
        out[0] = (float)r;
    }
}

extern "C" void kernel_launch(void* const* d_in, const int* in_sizes, int n_in,
                              void* d_out, int out_size, void* d_ws, size_t ws_size,
                              hipStream_t stream) {
    const float* x = (const float*)d_in[0];
    const float* y = (const float*)d_in[1];
    float* out = (float*)d_out;

    float*  base = (float*)d_ws;
    double* sab  = (double*)d_ws;
    float*  Rx   = base + 4;
    float*  Ry   = base + 4 + N_ROWS;
    float*  sqx  = base + 4 + 2 * N_ROWS;
    float*  sqy  = base + 4 + 3 * N_ROWS;

    (void)in_sizes; (void)n_in; (void)out_size; (void)ws_size;

    // 1) zero accumulators (Sab + Rx + Ry)
    dcov_zero<<<(4 + 2 * N_ROWS + 255) / 256, 256, 0, stream>>>(base);

    // 2) squared row norms
    row_sqnorm<<<N_ROWS / 8, 256, 0, stream>>>(x, sqx, DX);
    row_sqnorm<<<N_ROWS / 8, 256, 0, stream>>>(y, sqy, DY);

    // 3) fused WMMA Gram + distance + reductions
    //    each wave: 32x32 super-tile; block = 8 waves -> 256 cols
    dim3 grid(N_ROWS / 256, N_ROWS / 32);
    dcov_tile<<<grid, 256, 0, stream>>>(x, y, sqx, sqy, Rx, Ry, sab);

    // 4) final scalar combine
    dcov_final<<<1, 256, 0, stream>>>(Rx, Ry, sab, out);
}